// Conv2dSublayer_20426864460054
// MI455X (gfx1250) — compile-verified
//
#include <hip/hip_runtime.h>
#include <cstdint>

// out[n,c] = conv3x3(x[n,c], w[2c]) + conv3x3(x[n,c^4], w[2(c^4)+1])
// One block per (batch, channel-pair (c, c+4)): both input planes are pulled
// from HBM exactly once via CDNA5 async global->LDS DMA, staged with a 1-px
// halo, and feed both output channels (4-wide quads, 144 FMAs / 36 LDS floats
// per 8 outputs). Memory-bound: ~206 MB total traffic -> ~8.8 us floor @23.3TB/s.

typedef float v4f __attribute__((ext_vector_type(4)));

namespace {
constexpr int WDIM    = 56;
constexpr int PIX     = 56 * 56;            // 3136
constexpr int LSTRIDE = 64;                 // floats per LDS row (256 B)
constexpr int PLANE_F = 58 * LSTRIDE;       // 3712 floats / plane
constexpr int PLANE_B = PLANE_F * 4;        // 14848 bytes / plane
constexpr int COL0    = 4;                  // interior col offset (16B aligned)
}

__global__ __launch_bounds__(256)
void dwconv_butterfly_kernel(const float* __restrict__ x,
                             const float* __restrict__ w,
                             float* __restrict__ out)
{
    __shared__ float tile[2 * PLANE_F];     // 29,696 bytes

    const int tid = threadIdx.x;
    const int n   = blockIdx.y;                   // batch 0..31
    const int q   = blockIdx.x;                   // pair 0..127
    const int c   = ((q >> 2) << 3) | (q & 3);    // c with c%8 in 0..3
    const int cp  = c + 4;                        // partner = c ^ 4

    const int base0 = (n * 256 + c ) * PIX;       // element offsets
    const int base1 = (n * 256 + cp) * PIX;

    // ---- 1) issue async global->LDS DMA for both 56x56 interiors ----------
    // 14 x 16B segments tile one 224B row exactly -> div-free (rowid, seg)
    // mapping: 112 row-copies (2 planes * 56 rows), 16 lanes per row (14 live).
    const uint64_t sbase = (uint64_t)(uintptr_t)x;
    const unsigned lbase = (unsigned)(uintptr_t)&tile[0];
    const int      seg   = tid & 15;              // 0..15, live if < 14
    if (seg < 14) {
        #pragma unroll
        for (int it = 0; it < 7; ++it) {
            const int rowid = (tid >> 4) + it * 16;   // 0..111
            const int p     = rowid >= 56;
            const int row   = p ? rowid - 56 : rowid;
            const unsigned voff = (unsigned)((p ? base1 : base0) * 4
                                             + row * 224 + seg * 16);
            const unsigned lptr = lbase + (unsigned)(p * PLANE_B
                                             + (row + 1) * 256 + 16 + seg * 16);
            asm volatile("global_load_async_to_lds_b128 %0, %1, %2"
                         :: "v"(lptr), "v"(voff), "s"(sbase)
                         : "memory");
        }
    }

    // ---- 2) zero halo borders while the DMA is in flight ------------------
    if (tid < 64) {                 // rows 0 and 57: 2 planes * 2 rows * 16 b128
        const int p   = tid >> 5;
        const int r57 = (tid >> 4) & 1;
        const int sg  = tid & 15;
        const v4f z = {0.f, 0.f, 0.f, 0.f};
        *(v4f*)&tile[p * PLANE_F + (r57 ? 57 * LSTRIDE : 0) + sg * 4] = z;
    }
    if (tid < 224) {                // cols 3 and 60, rows 1..56, both planes
        const int p   = tid / 112;
        const int r   = tid % 112;
        const int col = (r < 56) ? (COL0 - 1) : (COL0 + 56);
        const int row = (r < 56 ? r : r - 56) + 1;
        tile[p * PLANE_F + row * LSTRIDE + col] = 0.0f;
    }

    // ---- 3) per-block filters (loads overlap the DMA too) -----------------
    // out c  : plane0 * w[2c]   + plane1 * w[2cp+1]
    // out cp : plane0 * w[2c+1] + plane1 * w[2cp]
    float wa[9], wb[9], wcc[9], wd[9];
    {
        const float* pw0 = w + 2 * c  * 9;   // filters 2c, 2c+1
        const float* pw1 = w + 2 * cp * 9;   // filters 2cp, 2cp+1
        #pragma unroll
        for (int k = 0; k < 9; ++k) {
            wa [k] = pw0[k];
            wb [k] = pw0[9 + k];
            wcc[k] = pw1[k];
            wd [k] = pw1[9 + k];
        }
    }

    // ---- 4) wait for this wave's DMA, then workgroup-sync -----------------
#if __has_builtin(__builtin_amdgcn_s_wait_asynccnt)
    __builtin_amdgcn_s_wait_asynccnt(0);
#else
    asm volatile("s_wait_asynccnt 0" ::: "memory");
#endif
    __syncthreads();

    // ---- 5) compute: 4-wide quads, both channels per iteration ------------
    // quad index i = row*14 + sg advances by 256 = 18*14 + 4 per iteration:
    // incremental (row, sg) update instead of div/mod in the hot loop.
    float* out0 = out + base0;
    float* out1 = out + base1;

    int row = tid / 14;              // one-time division at entry
    int sg  = tid - row * 14;
    for (int it = 0; it < 4; ++it) {
        if (row < 56) {
            const int x0 = sg * 4;

            float r0[3][6], r1[3][6];
            #pragma unroll
            for (int dy = 0; dy < 3; ++dy) {
                const int o = (row + dy) * LSTRIDE + (COL0 - 1) + x0;
                #pragma unroll
                for (int k = 0; k < 6; ++k) {
                    r0[dy][k] = tile[o + k];
                    r1[dy][k] = tile[PLANE_F + o + k];
                }
            }

            float a0[4] = {0.f, 0.f, 0.f, 0.f};
            float a1[4] = {0.f, 0.f, 0.f, 0.f};
            #pragma unroll
            for (int dy = 0; dy < 3; ++dy) {
                #pragma unroll
                for (int dx = 0; dx < 3; ++dx) {
                    const float fa = wa [dy * 3 + dx];
                    const float fb = wb [dy * 3 + dx];
                    const float fc = wcc[dy * 3 + dx];
                    const float fd = wd [dy * 3 + dx];
                    #pragma unroll
                    for (int j = 0; j < 4; ++j) {
                        const float p0 = r0[dy][dx + j];
                        const float p1 = r1[dy][dx + j];
                        a0[j] = fmaf(p0, fa, a0[j]);
                        a0[j] = fmaf(p1, fd, a0[j]);
                        a1[j] = fmaf(p0, fb, a1[j]);
                        a1[j] = fmaf(p1, fc, a1[j]);
                    }
                }
            }

            const v4f v0 = {a0[0], a0[1], a0[2], a0[3]};
            const v4f v1 = {a1[0], a1[1], a1[2], a1[3]};
            // streaming stores: output is written once, never re-read
            __builtin_nontemporal_store(v0, (v4f*)(out0 + row * WDIM + x0));
            __builtin_nontemporal_store(v1, (v4f*)(out1 + row * WDIM + x0));
        }
        // advance quad index by 256 (= 18*14 + 4)
        sg  += 4;
        row += 18;
        if (sg >= 14) { sg -= 14; row += 1; }
    }
}

extern "C" void kernel_launch(void* const* d_in, const int* in_sizes, int n_in,
                              void* d_out, int out_size, void* d_ws, size_t ws_size,
                              hipStream_t stream) {
    (void)in_sizes; (void)n_in; (void)out_size; (void)d_ws; (void)ws_size;
    const float* x = (const float*)d_in[0];   // (32, 256, 56, 56) fp32
    const float* w = (const float*)d_in[1];   // (512, 1, 3, 3)   fp32
    float* out = (float*)d_out;               // (32, 256, 56, 56) fp32

    dim3 grid(128, 32);                       // (channel pair, batch)
    dwconv_butterfly_kernel<<<grid, 256, 0, stream>>>(x, w, out);
}